// TopKRouter_81733227643376
// MI455X (gfx1250) — compile-verified
//
#include <hip/hip_runtime.h>

typedef __attribute__((ext_vector_type(2))) float v2f;
typedef __attribute__((ext_vector_type(8))) float v8f;

#define DIM   2048
#define NEXP  64
#define NTOK  32768          // 4 * 8192
#define KC    128            // k-chunk of W staged in LDS
#define WROW  (KC + 4)       // padded LDS row stride (floats): 132 -> bank-conflict-free
#define MBLK  128            // tokens per workgroup (8 waves * 16)

#if __has_builtin(__builtin_amdgcn_wmma_f32_16x16x4_f32)
#define HAVE_WMMA_F32X4 1
#else
#define HAVE_WMMA_F32X4 0
#endif

// ---------------------------------------------------------------------------
// Kernel 1: logits = x @ W^T  via V_WMMA_F32_16X16X4_F32 (exact f32 math).
// Each wave owns 16 tokens and all 4 expert tiles (E=64). W chunk in LDS.
// ---------------------------------------------------------------------------
__global__ __launch_bounds__(256) void gemm_logits_kernel(
    const float* __restrict__ x,   // [NTOK, DIM]
    const float* __restrict__ W,   // [NEXP, DIM]
    float* __restrict__ logits)    // [NTOK, NEXP]  (gate region of d_out)
{
    __shared__ float ldsW[NEXP * WROW];   // 64 * 132 * 4B = 33,792 B

    const int tid  = threadIdx.x;
    const int wave = tid >> 5;
    const int lane = tid & 31;
    const int half = lane >> 4;   // 0: K-pair {0,1}, 1: K-pair {2,3}
    const int l16  = lane & 15;
    const int m0   = blockIdx.x * MBLK + wave * 16;

    v8f acc0 = {}, acc1 = {}, acc2 = {}, acc3 = {};

    // A fragment source: lane (0..15) -> row m0+l16, columns k + half*2 .. +1
    const float* xrow = x + (size_t)(m0 + l16) * DIM + half * 2;

    for (int kb = 0; kb < DIM; kb += KC) {
        // ---- stage W[0:64, kb:kb+KC] into LDS, coalesced float4 ----
        for (int i = tid; i < NEXP * (KC / 4); i += 256) {
            const int e  = i >> 5;        // KC/4 = 32 float4 per expert row
            const int c4 = i & 31;
            const float4 v = *(const float4*)(W + (size_t)e * DIM + kb + c4 * 4);
            *(float4*)(&ldsW[e * WROW + c4 * 4]) = v;
        }
        __syncthreads();

        // ---- 32 WMMA k-steps over this chunk ----
#pragma unroll 4
        for (int k = 0; k < KC; k += 4) {
            const v2f a  = *(const v2f*)(xrow + kb + k);
            const int bo = k + half * 2;
            const v2f b0 = *(const v2f*)(&ldsW[(0 * 16 + l16) * WROW + bo]);
            const v2f b1 = *(const v2f*)(&ldsW[(1 * 16 + l16) * WROW + bo]);
            const v2f b2 = *(const v2f*)(&ldsW[(2 * 16 + l16) * WROW + bo]);
            const v2f b3 = *(const v2f*)(&ldsW[(3 * 16 + l16) * WROW + bo]);
#if HAVE_WMMA_F32X4
            acc0 = __builtin_amdgcn_wmma_f32_16x16x4_f32(false, a, false, b0, (short)0, acc0, false, false);
            acc1 = __builtin_amdgcn_wmma_f32_16x16x4_f32(false, a, false, b1, (short)0, acc1, false, false);
            acc2 = __builtin_amdgcn_wmma_f32_16x16x4_f32(false, a, false, b2, (short)0, acc2, false, false);
            acc3 = __builtin_amdgcn_wmma_f32_16x16x4_f32(false, a, false, b3, (short)0, acc3, false, false);
#else
            // Fallback (should not trigger on gfx1250): scalar FMA so it still compiles.
            acc0[0] += a.x * b0.x + a.y * b0.y;
            acc1[0] += a.x * b1.x + a.y * b1.y;
            acc2[0] += a.x * b2.x + a.y * b2.y;
            acc3[0] += a.x * b3.x + a.y * b3.y;
#endif
        }
        __syncthreads();
    }

    // ---- store: C/D layout: VGPR r -> token m0 + r + 8*half, expert et*16 + l16 ----
#pragma unroll
    for (int r = 0; r < 8; ++r) {
        const int tok = m0 + r + 8 * half;
        float* o = logits + (size_t)tok * NEXP;
        o[0 * 16 + l16] = acc0[r];
        o[1 * 16 + l16] = acc1[r];
        o[2 * 16 + l16] = acc2[r];
        o[3 * 16 + l16] = acc3[r];
    }
}

// ---------------------------------------------------------------------------
// Kernel 2: per-row bias add + softmax (in place) + top-2 (value,index).
// One wave32 per token row; 2 experts per lane.
// ---------------------------------------------------------------------------
__global__ __launch_bounds__(256) void softmax_top2_kernel(
    const float* __restrict__ bias,   // [NEXP]
    float* __restrict__ gates,        // in: logits, out: softmax probs [NTOK, NEXP]
    float* __restrict__ topk_idx_f,   // [NTOK, 2] (indices written as float values)
    float* __restrict__ topk_val)     // [NTOK, 2]
{
    const int lane = threadIdx.x & 31;
    const int wave = threadIdx.x >> 5;
    const int row  = blockIdx.x * 8 + wave;

    float* g = gates + (size_t)row * NEXP;
    const v2f v  = *(const v2f*)(g + 2 * lane);
    const v2f bb = *(const v2f*)(bias + 2 * lane);
    const float x0 = v.x + bb.x;
    const float x1 = v.y + bb.y;

    // max over 64
    float m = fmaxf(x0, x1);
#pragma unroll
    for (int off = 16; off > 0; off >>= 1)
        m = fmaxf(m, __shfl_xor(m, off, 32));

    const float e0 = __expf(x0 - m);
    const float e1 = __expf(x1 - m);
    float s = e0 + e1;
#pragma unroll
    for (int off = 16; off > 0; off >>= 1)
        s += __shfl_xor(s, off, 32);
    const float inv = 1.0f / s;
    const float p0 = e0 * inv;
    const float p1 = e1 * inv;

    v2f pv; pv.x = p0; pv.y = p1;
    *(v2f*)(g + 2 * lane) = pv;          // in-place: each lane owns its 2 slots

    // lane-local top-2 (lower index wins ties, matching jax.lax.top_k)
    float a1, a2; int i1, i2;
    if (p0 >= p1) { a1 = p0; i1 = 2 * lane;     a2 = p1; i2 = 2 * lane + 1; }
    else          { a1 = p1; i1 = 2 * lane + 1; a2 = p0; i2 = 2 * lane;     }

#pragma unroll
    for (int off = 16; off > 0; off >>= 1) {
        const float b1v = __shfl_xor(a1, off, 32);
        const int   b1i = __shfl_xor(i1, off, 32);
        const float b2v = __shfl_xor(a2, off, 32);
        const int   b2i = __shfl_xor(i2, off, 32);
        // merge top-2 of {a1,a2} and {b1v,b2v}
        const bool swp = (b1v > a1) || (b1v == a1 && b1i < i1);
        const float t1v = swp ? b1v : a1;  const int t1i = swp ? b1i : i1;
        const float c1v = swp ? a1  : b1v; const int c1i = swp ? i1  : b1i; // loser of firsts
        const float c2v = swp ? b2v : a2;  const int c2i = swp ? b2i : i2;  // winner's second
        const bool s2 = (c2v > c1v) || (c2v == c1v && c2i < c1i);
        a1 = t1v; i1 = t1i;
        a2 = s2 ? c2v : c1v;
        i2 = s2 ? c2i : c1i;
    }

    if (lane == 0) {
        topk_idx_f[row * 2 + 0] = (float)i1;
        topk_idx_f[row * 2 + 1] = (float)i2;
        topk_val[row * 2 + 0]   = a1;
        topk_val[row * 2 + 1]   = a2;
    }
}

extern "C" void kernel_launch(void* const* d_in, const int* in_sizes, int n_in,
                              void* d_out, int out_size, void* d_ws, size_t ws_size,
                              hipStream_t stream) {
    const float* x = (const float*)d_in[0];   // [4, 8192, 2048]
    const float* W = (const float*)d_in[1];   // [64, 2048]
    const float* b = (const float*)d_in[2];   // [64]

    float* out    = (float*)d_out;
    float* gates  = out;                                  // [NTOK, 64]
    float* idxf   = out + (size_t)NTOK * NEXP;            // [NTOK, 2]
    float* tval   = idxf + (size_t)NTOK * 2;              // [NTOK, 2]

    gemm_logits_kernel<<<NTOK / MBLK, 256, 0, stream>>>(x, W, gates);
    softmax_top2_kernel<<<NTOK / 8, 256, 0, stream>>>(b, gates, idxf, tval);
}